// PhysicsModel_46823733461271
// MI455X (gfx1250) — compile-verified
//
#include <hip/hip_runtime.h>
#include <stdint.h>

// Problem geometry (fixed by the reference)
#define BSZ   8192
#define TLEN  2048
#define NC    16                 // chunks along T
#define CL    128                // chunk length (= 2^7, so A^CL via 7 squarings)
#define ROWS  32                 // rows per wave (= wave32 lanes)
#define RSTR  (CL + 4)           // LDS row stride in floats: 528 B, 16B aligned
#define WPB   2                  // waves per block
#define TPB   (WPB * 32)
#define RB    (BSZ / ROWS)       // 256 row-blocks
#define NWAVE (RB * NC)          // 4096 waves per pass

// ---- gfx1250 async LDS<->global helpers (ASYNCcnt path) --------------------
__device__ __forceinline__ unsigned lds_off(const void* p) {
    // generic pointer to LDS: low 32 bits == workgroup-relative LDS byte offset
    return (unsigned)(unsigned long long)p;
}
__device__ __forceinline__ void async_load_b128(unsigned lds, const float* g) {
    asm volatile("global_load_async_to_lds_b128 %0, %1, off"
                 :: "v"(lds), "v"((unsigned long long)g) : "memory");
}
__device__ __forceinline__ void async_store_b128(float* g, unsigned lds) {
    asm volatile("global_store_async_from_lds_b128 %0, %1, off"
                 :: "v"((unsigned long long)g), "v"(lds) : "memory");
}
__device__ __forceinline__ void wait_async0() {
    asm volatile("s_wait_asynccnt 0" ::: "memory");
}
__device__ __forceinline__ void wait_ds0() {
    asm volatile("s_wait_dscnt 0" ::: "memory");
}

// ---- Pass 1: per (rowblock, chunk) boundary vectors from zero state --------
__global__ __launch_bounds__(TPB) void physics_pass1(
    const float* __restrict__ u,
    const float* dtp, const float* mp, const float* cp, const float* kp,
    const float* imp, const float* isp,
    float2* __restrict__ bvec)          // layout [chunk][row]
{
    __shared__ float tile[WPB * ROWS * RSTR];
    const int lane = threadIdx.x & 31;
    const int wid  = threadIdx.x >> 5;
    const int wave = blockIdx.x * WPB + wid;
    const int rb   = wave & (RB - 1);
    const int j    = wave >> 8;         // log2(RB) == 8
    const int row0 = rb * ROWS;
    const int t0   = j * CL;

    float* slab = &tile[wid * ROWS * RSTR];
    const unsigned l0 = lds_off(slab);
    const float* gbase = u + (size_t)row0 * TLEN + t0;
    #pragma unroll
    for (int r = 0; r < ROWS; ++r)      // one coalesced 512B line per row
        async_load_b128(l0 + (unsigned)(r * RSTR * 4 + lane * 16),
                        gbase + (size_t)r * TLEN + lane * 4);
    wait_async0();

    const float DT = *dtp, IM = *imp, IS = *isp;
    const float Cc = *cp, Kk = *kp, invM = 1.0f / *mp;   // m == 1.0 -> exact
    const float* rowp = slab + lane * RSTR;
    float x = 0.0f, v = 0.0f;
    #pragma unroll 8
    for (int t = 0; t < CL; ++t) {
        float up  = rowp[t] * IS + IM;
        float acc = (up - Cc * v - Kk * x) * invM;
        float nx  = x + DT * v;
        float nv  = v + DT * acc;
        x = nx; v = nv;
    }
    bvec[(size_t)j * BSZ + row0 + lane] = make_float2(x, v);
}

// ---- Pass 2: in-register chunk scan + replay + y/state outputs -------------
__global__ __launch_bounds__(TPB) void physics_pass2(
    const float* __restrict__ u,
    float* __restrict__ out,            // B*T ys then B*2 final states
    const float2* __restrict__ bvec,
    const float* dtp, const float* mp, const float* cp, const float* kp,
    const float* imp, const float* isp, const float* omp, const float* osp)
{
    __shared__ float tile[WPB * ROWS * RSTR];
    const int lane = threadIdx.x & 31;
    const int wid  = threadIdx.x >> 5;
    const int wave = blockIdx.x * WPB + wid;
    const int rb   = wave & (RB - 1);
    const int j    = wave >> 8;
    const int row0 = rb * ROWS;
    const int t0   = j * CL;
    const int row  = row0 + lane;

    float* slab = &tile[wid * ROWS * RSTR];
    const unsigned l0 = lds_off(slab);
    const float* gbase = u + (size_t)row0 * TLEN + t0;
    #pragma unroll
    for (int r = 0; r < ROWS; ++r)
        async_load_b128(l0 + (unsigned)(r * RSTR * 4 + lane * 16),
                        gbase + (size_t)r * TLEN + lane * 4);

    // Overlap with the async loads: recover this chunk's true start state.
    const float DT = *dtp, IM = *imp, IS = *isp;
    const float Cc = *cp, Kk = *kp, invM = 1.0f / *mp;
    const float OM = *omp, invOS = 1.0f / *osp;
    double sx = 0.0, sv = 0.0;
    if (j > 0) {
        const double dt = (double)DT, m = (double)*mp;
        double a00 = 1.0,                  a01 = dt;
        double a10 = -(double)Kk * dt / m, a11 = 1.0 - (double)Cc * dt / m;
        #pragma unroll
        for (int i = 0; i < 7; ++i) {     // A^(2^7) = A^CL
            double n00 = a00*a00 + a01*a10, n01 = a00*a01 + a01*a11;
            double n10 = a10*a00 + a11*a10, n11 = a10*a01 + a11*a11;
            a00 = n00; a01 = n01; a10 = n10; a11 = n11;
        }
        for (int jj = 0; jj < j; ++jj) {  // s_{jj+1} = A^CL * s_jj + b_jj
            float2 b = bvec[(size_t)jj * BSZ + row];
            double nx = a00*sx + a01*sv + (double)b.x;
            double nv = a10*sx + a11*sv + (double)b.y;
            sx = nx; sv = nv;
        }
    }
    float x = (float)sx, v = (float)sv;

    wait_async0();
    float* rowp = slab + lane * RSTR;
    #pragma unroll 8
    for (int t = 0; t < CL; ++t) {
        float up  = rowp[t] * IS + IM;
        float acc = (up - Cc * v - Kk * x) * invM;
        float nx  = x + DT * v;
        float nv  = v + DT * acc;
        float ay  = (up - Cc * nv - Kk * nx) * invM;
        rowp[t]   = (ay - OM) * invOS;    // overwrite u tile with y in place
        x = nx; v = nv;
    }
    wait_ds0();                           // LDS y-tile visible to async engine

    float* obase = out + (size_t)row0 * TLEN + t0;
    #pragma unroll
    for (int r = 0; r < ROWS; ++r)
        async_store_b128(obase + (size_t)r * TLEN + lane * 4,
                         l0 + (unsigned)(r * RSTR * 4 + lane * 16));

    if (j == NC - 1)                      // final states live at d_out + B*T
        ((float2*)(out + (size_t)BSZ * TLEN))[row] = make_float2(x, v);

    wait_async0();
}

extern "C" void kernel_launch(void* const* d_in, const int* in_sizes, int n_in,
                              void* d_out, int out_size, void* d_ws, size_t ws_size,
                              hipStream_t stream)
{
    (void)in_sizes; (void)n_in; (void)out_size; (void)ws_size;
    const float* u  = (const float*)d_in[0];
    const float* dt = (const float*)d_in[1];
    const float* m  = (const float*)d_in[2];
    const float* c  = (const float*)d_in[3];
    const float* k  = (const float*)d_in[4];
    const float* im = (const float*)d_in[5];
    const float* is = (const float*)d_in[6];
    const float* om = (const float*)d_in[7];
    const float* os = (const float*)d_in[8];
    float2* bvec = (float2*)d_ws;        // NC*B*8 B = 1 MiB scratch

    dim3 blk(TPB);
    dim3 grd(NWAVE / WPB);               // 2048 blocks x 2 waves
    physics_pass1<<<grd, blk, 0, stream>>>(u, dt, m, c, k, im, is, bvec);
    physics_pass2<<<grd, blk, 0, stream>>>(u, (float*)d_out, bvec,
                                           dt, m, c, k, im, is, om, os);
}